// PNABaseModel_44573170597949
// MI455X (gfx1250) — compile-verified
//
#include <hip/hip_runtime.h>
#include <math.h>

// ---------------- problem constants ----------------
static constexpr int kN      = 25000;   // nodes
static constexpr int kE      = 400000;  // edges
static constexpr int kD      = 64;      // feature dim
static constexpr int kL      = 2;       // layers
static constexpr int kNE     = 4;       // edge types
static constexpr int kB      = 64;      // graphs
static constexpr int kMAXDEG = 64;
static constexpr float kEPS  = 1e-5f;

// ---------------- WMMA helpers (CDNA5 / gfx1250, wave32) ----------------
typedef __attribute__((ext_vector_type(2))) float v2f;
typedef __attribute__((ext_vector_type(8))) float v8f;

static __device__ __forceinline__ v8f wmma4(v2f a, v2f b, v8f c) {
  // D = A(16x4 f32) * B(4x16 f32) + C(16x16 f32)
  return __builtin_amdgcn_wmma_f32_16x16x4_f32(
      /*neg_a=*/false, a, /*neg_b=*/false, b,
      /*c_mod=*/(short)0, c, /*reuse_a=*/false, /*reuse_b=*/false);
}

// ---------------- tiny utility kernels ----------------
__global__ void k_fill_u32(unsigned* __restrict__ p, unsigned v, long n) {
  long i = (long)blockIdx.x * blockDim.x + threadIdx.x;
  if (i < n) p[i] = v;
}

__global__ void k_copy_f4(const float4* __restrict__ src, float4* __restrict__ dst, long n4) {
  long i = (long)blockIdx.x * blockDim.x + threadIdx.x;
  if (i < n4) dst[i] = src[i];
}

// delta = sum(deg_hist * log(bin+1)) / sum(deg_hist)
__global__ void k_delta(const float* __restrict__ deg_hist, float* __restrict__ delta) {
  __shared__ float s1[kMAXDEG];
  __shared__ float s2[kMAXDEG];
  int j = threadIdx.x;
  float w = deg_hist[j];
  s1[j] = w * logf((float)j + 1.0f);
  s2[j] = w;
  __syncthreads();
  for (int off = kMAXDEG >> 1; off > 0; off >>= 1) {
    if (j < off) { s1[j] += s1[j + off]; s2[j] += s2[j + off]; }
    __syncthreads();
  }
  if (j == 0) delta[0] = s1[0] / s2[0];
}

// in-degree via float atomics on dst
__global__ void k_deg(const int* __restrict__ edge_index, float* __restrict__ deg) {
  int e = blockIdx.x * blockDim.x + threadIdx.x;
  if (e < kE) atomicAdd(&deg[edge_index[kE + e]], 1.0f);
}

// amp / att scalers per node
__global__ void k_scalers(const float* __restrict__ deg, const float* __restrict__ pdelta,
                          float* __restrict__ amp, float* __restrict__ att) {
  int i = blockIdx.x * blockDim.x + threadIdx.x;
  if (i >= kN) return;
  float d     = deg[i];
  float delta = pdelta[0];
  amp[i] = logf(d + 1.0f) / delta;
  att[i] = delta / logf(fmaxf(d, 1.0f) + 1.0f);
}

// T[t] = edge_emb[t] @ W_pre[128:192] + b_pre   (per edge-type bias vector)
__global__ void k_edge_type(const float* __restrict__ emb, const float* __restrict__ Wp,
                            const float* __restrict__ b_pre, float* __restrict__ T) {
  int t = blockIdx.x;       // 0..3
  int j = threadIdx.x;      // 0..63
  float acc = b_pre[j];
  #pragma unroll 8
  for (int k = 0; k < kD; ++k)
    acc += emb[t * kD + k] * Wp[(128 + k) * kD + j];
  T[t * kD + j] = acc;
}

// ---------------- GEMM 1: AB[N,128] = X[N,64] @ [W0 | W1] ----------------
// grid = (ceil(N/16), 8), block = 32 (one wave per 16x16 output tile)
__global__ void k_pre_gemm(const float* __restrict__ X, const float* __restrict__ Wp,
                           float* __restrict__ AB) {
  const int lane = threadIdx.x;
  const int m0   = blockIdx.x * 16;
  const int n0   = blockIdx.y * 16;          // 0..112
  const int half = lane >> 4;                // 0: K pair {k,k+1}, 1: {k+2,k+3}
  const int l15  = lane & 15;
  int row = m0 + l15; if (row >= kN) row = kN - 1;   // clamped dup-load, store guarded
  const int n      = n0 + l15;
  const int rowOff = (n0 >= kD) ? kD : 0;    // W0 rows 0..63, W1 rows 64..127
  const int nb     = n & (kD - 1);

  v8f acc = {};
  #pragma unroll
  for (int k = 0; k < kD; k += 4) {
    const int bk = k + half * 2;
    v2f a = *(const v2f*)(X + row * kD + bk);
    v2f b;
    b.x = Wp[(rowOff + bk    ) * kD + nb];
    b.y = Wp[(rowOff + bk + 1) * kD + nb];
    acc = wmma4(a, b, acc);
  }
  #pragma unroll
  for (int v = 0; v < 8; ++v) {
    int m = m0 + v + half * 8;
    if (m < kN) AB[m * 128 + n] = acc[v];
  }
}

// ---------------- fused edge message + scatter aggregation ----------------
// 64 threads per edge (channel j), 4 edges per 256-thread block.
// m = relu(AB[dst][j] + AB[src][64+j] + T[attr][j]) >= 0
// sum/sumsq: f32 atomic add; max/min: monotone unsigned-bit atomics (m >= 0).
__global__ void k_edge_scatter(const int* __restrict__ edge_index, const int* __restrict__ edge_attr,
                               const float* __restrict__ AB, const float* __restrict__ T,
                               float* __restrict__ sum, float* __restrict__ sumsq,
                               unsigned* __restrict__ mx, unsigned* __restrict__ mn) {
  const int e = blockIdx.x * 4 + (threadIdx.x >> 6);
  const int j = threadIdx.x & 63;
  if (e >= kE) return;
  const int s = edge_index[e];
  const int d = edge_index[kE + e];
  const int a = edge_attr[e];
  float m = AB[d * 128 + j] + AB[s * 128 + kD + j] + T[a * kD + j];
  m = fmaxf(m, 0.0f);
  const int o = d * kD + j;
  atomicAdd(&sum[o],   m);
  atomicAdd(&sumsq[o], m * m);
  atomicMax(&mx[o], __float_as_uint(m));
  atomicMin(&mn[o], __float_as_uint(m));
}

// finalize in place: sum->mean, sumsq->std, mx/mn -> 0 for degree-0 nodes
__global__ void k_finalize(const float* __restrict__ deg,
                           float* __restrict__ sum, float* __restrict__ sumsq,
                           float* __restrict__ mx, float* __restrict__ mn) {
  long idx = (long)blockIdx.x * blockDim.x + threadIdx.x;
  if (idx >= (long)kN * kD) return;
  int i = (int)(idx >> 6);
  float d    = deg[i];
  float dc   = fmaxf(d, 1.0f);
  float mean = sum[idx] / dc;
  float sq   = sumsq[idx] / dc;
  float sd   = sqrtf(fmaxf(sq - mean * mean, 0.0f) + kEPS);
  sum[idx]   = mean;
  sumsq[idx] = sd;
  if (d <= 0.0f) { mx[idx] = 0.0f; mn[idx] = 0.0f; }
}

// ---------------- GEMM 2: post-mix with fused PNA scalers + residual ----------------
// aggr(m, k): k in [0,64)->mean, [64,128)->max, [128,192)->min, [192,256)->std
// X += (aggr@Wq0 + amp*(aggr@Wq1) + att*(aggr@Wq2) + b_post) * snorm
// grid = (ceil(N/16), 4), block = 32
__global__ void k_post_gemm(const float* __restrict__ mean, const float* __restrict__ mxb,
                            const float* __restrict__ mnb, const float* __restrict__ stdb,
                            const float* __restrict__ Wq,   const float* __restrict__ b_post,
                            const float* __restrict__ amp,  const float* __restrict__ att,
                            const float* __restrict__ snorm, float* __restrict__ Xio) {
  const int lane = threadIdx.x;
  const int m0   = blockIdx.x * 16;
  const int n0   = blockIdx.y * 16;         // 0..48
  const int half = lane >> 4;
  const int l15  = lane & 15;
  int row = m0 + l15; if (row >= kN) row = kN - 1;
  const int n = n0 + l15;

  v8f acc0 = {}, acc1 = {}, acc2 = {};
  #pragma unroll
  for (int seg = 0; seg < 4; ++seg) {
    const float* ap = (seg == 0) ? mean : (seg == 1) ? mxb : (seg == 2) ? mnb : stdb;
    #pragma unroll
    for (int k = 0; k < kD; k += 4) {
      const int bk = k + half * 2;          // within segment
      const int gk = seg * kD + bk;         // global K (row of W_post block)
      v2f a = *(const v2f*)(ap + row * kD + bk);
      v2f b0, b1, b2;
      b0.x = Wq[(gk      ) * kD + n]; b0.y = Wq[(gk + 1      ) * kD + n];
      b1.x = Wq[(gk + 256) * kD + n]; b1.y = Wq[(gk + 1 + 256) * kD + n];
      b2.x = Wq[(gk + 512) * kD + n]; b2.y = Wq[(gk + 1 + 512) * kD + n];
      acc0 = wmma4(a, b0, acc0);
      acc1 = wmma4(a, b1, acc1);
      acc2 = wmma4(a, b2, acc2);
    }
  }
  #pragma unroll
  for (int v = 0; v < 8; ++v) {
    int m = m0 + v + half * 8;
    if (m < kN) {
      float p = acc0[v] + amp[m] * acc1[v] + att[m] * acc2[v] + b_post[n];
      Xio[m * kD + n] += p * snorm[m];
    }
  }
}

// ---------------- readout ----------------
__global__ void k_pool(const float* __restrict__ X, const int* __restrict__ batch,
                       float* __restrict__ pooled) {
  long idx = (long)blockIdx.x * blockDim.x + threadIdx.x;
  if (idx >= (long)kN * kD) return;
  int i = (int)(idx >> 6);
  int j = (int)(idx & 63);
  atomicAdd(&pooled[batch[i] * kD + j], X[idx]);
}

__global__ void k_readout(const float* __restrict__ pooled,
                          const float* __restrict__ W_h, const float* __restrict__ b_h,
                          const float* __restrict__ W_o, const float* __restrict__ b_o,
                          float* __restrict__ y) {
  __shared__ float sp[kD];
  __shared__ float sr[kD];
  const int b = blockIdx.x;
  const int j = threadIdx.x;
  sp[j] = pooled[b * kD + j];
  __syncthreads();
  float h = b_h[j];
  #pragma unroll 8
  for (int k = 0; k < kD; ++k) h += sp[k] * W_h[k * kD + j];
  h = fmaxf(h, 0.0f);
  sr[j] = h * W_o[j];
  __syncthreads();
  for (int off = kD >> 1; off > 0; off >>= 1) {
    if (j < off) sr[j] += sr[j + off];
    __syncthreads();
  }
  if (j == 0) y[b] = sr[0] + b_o[0];
}

// ---------------- host orchestration ----------------
extern "C" void kernel_launch(void* const* d_in, const int* in_sizes, int n_in,
                              void* d_out, int out_size, void* d_ws, size_t ws_size,
                              hipStream_t stream) {
  (void)in_sizes; (void)n_in; (void)out_size; (void)ws_size;

  const float* X_n        = (const float*)d_in[0];
  const int*   edge_index = (const int*)  d_in[1];
  const int*   edge_attr  = (const int*)  d_in[2];
  const float* snorm      = (const float*)d_in[3];
  const int*   batch      = (const int*)  d_in[4];
  const float* deg_hist   = (const float*)d_in[5];
  const float* edge_emb   = (const float*)d_in[6];   // [L, NE, D]
  const float* W_pre      = (const float*)d_in[7];   // [L, 3D, D]
  const float* b_pre      = (const float*)d_in[8];   // [L, D]
  const float* W_post     = (const float*)d_in[9];   // [L, 12D, D]
  const float* b_post     = (const float*)d_in[10];  // [L, D]
  const float* W_h        = (const float*)d_in[11];
  const float* b_h        = (const float*)d_in[12];
  const float* W_o        = (const float*)d_in[13];
  const float* b_o        = (const float*)d_in[14];
  float*       y          = (float*)d_out;

  // carve workspace (256B aligned regions)
  char* w = (char*)d_ws;
  auto carve = [&](size_t bytes) -> char* {
    char* p = w;
    w += (bytes + 255) & ~(size_t)255;
    return p;
  };
  float* delta  = (float*)carve(sizeof(float));
  float* deg    = (float*)carve(sizeof(float) * kN);
  float* amp    = (float*)carve(sizeof(float) * kN);
  float* att    = (float*)carve(sizeof(float) * kN);
  float* Xcur   = (float*)carve(sizeof(float) * kN * kD);
  float* AB     = (float*)carve(sizeof(float) * kN * 128);
  // sum | sumsq | mx contiguous so one fill zeroes all three
  float* aggZ   = (float*)carve(sizeof(float) * kN * kD * 3);
  float* sum    = aggZ;
  float* sumsq  = aggZ + (size_t)kN * kD;
  float* mx     = aggZ + (size_t)2 * kN * kD;
  float* mn     = (float*)carve(sizeof(float) * kN * kD);
  float* T      = (float*)carve(sizeof(float) * kNE * kD);
  float* pooled = (float*)carve(sizeof(float) * kB * kD);

  const long nNodeCh = (long)kN * kD;
  const int  thr     = 256;
  auto blocks = [](long n, int t) { return (unsigned)((n + t - 1) / t); };

  // ---- global prep ----
  k_delta<<<1, kMAXDEG, 0, stream>>>(deg_hist, delta);
  k_fill_u32<<<blocks(kN, thr), thr, 0, stream>>>((unsigned*)deg, 0u, kN);
  k_fill_u32<<<blocks(kB * kD, thr), thr, 0, stream>>>((unsigned*)pooled, 0u, kB * kD);
  k_copy_f4<<<blocks(nNodeCh / 4, thr), thr, 0, stream>>>(
      (const float4*)X_n, (float4*)Xcur, nNodeCh / 4);
  k_deg<<<blocks(kE, thr), thr, 0, stream>>>(edge_index, deg);
  k_scalers<<<blocks(kN, thr), thr, 0, stream>>>(deg, delta, amp, att);

  const dim3 gemm_blk(32, 1, 1);
  const dim3 pre_grd((kN + 15) / 16, 8, 1);
  const dim3 post_grd((kN + 15) / 16, 4, 1);

  // ---- PNA layers ----
  for (int l = 0; l < kL; ++l) {
    const float* Wp  = W_pre  + (size_t)l * 3  * kD * kD;
    const float* bp  = b_pre  + (size_t)l * kD;
    const float* Wq  = W_post + (size_t)l * 12 * kD * kD;
    const float* bq  = b_post + (size_t)l * kD;
    const float* emb = edge_emb + (size_t)l * kNE * kD;

    // init aggregation buffers: sum/sumsq/mx = 0, mn = +FLT_MAX bits
    k_fill_u32<<<blocks(nNodeCh * 3, thr), thr, 0, stream>>>((unsigned*)aggZ, 0u, nNodeCh * 3);
    k_fill_u32<<<blocks(nNodeCh, thr), thr, 0, stream>>>((unsigned*)mn, 0x7F7FFFFFu, nNodeCh);

    // AB = X @ [W0 | W1]  (WMMA f32)
    k_pre_gemm<<<pre_grd, gemm_blk, 0, stream>>>(Xcur, Wp, AB);
    // per-edge-type bias vectors
    k_edge_type<<<kNE, kD, 0, stream>>>(emb, Wp, bp, T);
    // fused message + 4-way scatter aggregation
    k_edge_scatter<<<(kE + 3) / 4, 256, 0, stream>>>(
        edge_index, edge_attr, AB, T, sum, sumsq, (unsigned*)mx, (unsigned*)mn);
    // mean/std in place, mask deg-0 max/min
    k_finalize<<<blocks(nNodeCh, thr), thr, 0, stream>>>(deg, sum, sumsq, mx, mn);
    // X += (aggr@Wq0 + amp*aggr@Wq1 + att*aggr@Wq2 + b_post) * snorm  (WMMA f32)
    k_post_gemm<<<post_grd, gemm_blk, 0, stream>>>(
        sum, mx, mn, sumsq, Wq, bq, amp, att, snorm, Xcur);
  }

  // ---- readout ----
  k_pool<<<blocks(nNodeCh, thr), thr, 0, stream>>>(Xcur, batch, pooled);
  k_readout<<<kB, kD, 0, stream>>>(pooled, W_h, b_h, W_o, b_o, y);
}